// GCNEncoder_25202868093390
// MI455X (gfx1250) — compile-verified
//
#include <hip/hip_runtime.h>
#include <stdint.h>

typedef __attribute__((ext_vector_type(2))) float v2f;
typedef __attribute__((ext_vector_type(8))) float v8f;

#define KC 128            // K-chunk staged in LDS (all K here are multiples of 128)
#define PAIR_STRIDE 160   // floats per k-row-pair: 128 data + 32 pad
// Pair-interleaved LDS layout: element (k, c) lives at
//   (k/2)*PAIR_STRIDE + c*2 + (k&1)
// so a lane's B fragment {B[k][c], B[k+1][c]} is one aligned b64 load, and the
// +160 half-wave offset === 32 (mod 64 banks) makes b64 reads conflict-free.

// ---------------------------------------------------------------------------
// WMMA fp32 GEMM: C[M,N] = A[M,K] @ B[K,N] (+ bias).
// Call-site guarantees: N % 64 == 0, K % 128 == 0, M % 16 == 0.
// Block = 256 threads = 8 waves sharing one 64-wide B panel in LDS; wave w
// computes the 16x64 strip for M-tile (blockIdx.y*8 + w).
// Fragment layouts per CDNA5 ISA (wave32):
//   A 16x4:  lane L holds M=L%16;  vgpr v holds K = 2*(L/16) + v
//   B 4x16:  lane L holds N=L%16;  vgpr v holds K = 2*(L/16) + v
//   C 16x16: lane L holds N=L%16;  vgpr v holds M = v + 8*(L/16)
// ---------------------------------------------------------------------------
template <bool BIAS>
__global__ __launch_bounds__(256) void k_gemm_f32_wmma(
    const float* __restrict__ A, const float* __restrict__ B,
    const float* __restrict__ bias, float* __restrict__ C,
    int M, int K, int N)
{
    __shared__ float ldsB[(KC / 2) * PAIR_STRIDE];   // 40 KB

    const int lane  = threadIdx.x & 31;
    const int wib   = threadIdx.x >> 5;
    const int col0  = blockIdx.x << 6;                  // 64-column strip
    const int tileM = blockIdx.y * 8 + wib;
    const bool active = (tileM * 16) < M;
    const int row0  = active ? (tileM * 16) : 0;        // clamp: compute, don't store
    const int m16   = lane & 15;
    const int half  = lane >> 4;

    v8f acc0 = (v8f)0.0f, acc1 = (v8f)0.0f, acc2 = (v8f)0.0f, acc3 = (v8f)0.0f;

    const float* arow = A + (size_t)(row0 + m16) * K + 2 * half;

    for (int kc = 0; kc < K; kc += KC) {
        // ---- stage B[kc..kc+KC-1, col0..col0+63] into LDS (pair-interleaved) ----
        {
            const float* Bp = B + (size_t)kc * N + col0;
#pragma unroll
            for (int it = 0; it < (KC / 2) * 32 / 256; ++it) {   // 8 iterations
                int u  = it * 256 + threadIdx.x;   // unit = (row-pair p, col-pair c)
                int p  = u >> 5;                   // 0..63
                int c  = (u & 31) << 1;            // 0,2,..,62
                const float* r0 = Bp + (size_t)(2 * p) * N + c;
                float2 lo = *reinterpret_cast<const float2*>(r0);       // row 2p
                float2 hi = *reinterpret_cast<const float2*>(r0 + N);   // row 2p+1
                float4 out = make_float4(lo.x, hi.x, lo.y, hi.y);
                *reinterpret_cast<float4*>(&ldsB[p * PAIR_STRIDE + c * 2]) = out;
            }
        }
        __syncthreads();

        const float* ac = arow + kc;
#pragma unroll 4
        for (int k0 = 0; k0 < KC; k0 += 4) {
            v2f a;
            a.x = ac[k0 + 0];
            a.y = ac[k0 + 1];
            // lane's fragment base for this k-step: pair (k0/2 + half), col m16
            const float* bp = ldsB + (k0 >> 1) * PAIR_STRIDE
                                   + half * PAIR_STRIDE + m16 * 2;
            v2f b0 = *reinterpret_cast<const v2f*>(bp);        // cols +0
            v2f b1 = *reinterpret_cast<const v2f*>(bp + 32);   // cols +16
            v2f b2 = *reinterpret_cast<const v2f*>(bp + 64);   // cols +32
            v2f b3 = *reinterpret_cast<const v2f*>(bp + 96);   // cols +48
            acc0 = __builtin_amdgcn_wmma_f32_16x16x4_f32(false, a, false, b0,
                                                         (short)0, acc0, false, false);
            acc1 = __builtin_amdgcn_wmma_f32_16x16x4_f32(false, a, false, b1,
                                                         (short)0, acc1, false, false);
            acc2 = __builtin_amdgcn_wmma_f32_16x16x4_f32(false, a, false, b2,
                                                         (short)0, acc2, false, false);
            acc3 = __builtin_amdgcn_wmma_f32_16x16x4_f32(false, a, false, b3,
                                                         (short)0, acc3, false, false);
        }
        __syncthreads();
    }

    if (!active) return;   // wave-uniform; all barriers already passed

    v8f accs[4] = {acc0, acc1, acc2, acc3};
#pragma unroll
    for (int t = 0; t < 4; ++t) {
        const int c    = col0 + t * 16 + m16;
        const float bv = BIAS ? bias[c] : 0.0f;
#pragma unroll
        for (int v = 0; v < 8; ++v) {
            const int r = row0 + v + 8 * half;
            C[(size_t)r * N + c] = accs[t][v] + bv;
        }
    }
}

// ---------------------------------------------------------------------------
// Elementwise / graph kernels
// ---------------------------------------------------------------------------
__global__ void k_fill(float* __restrict__ p, int n, float v)
{
    int i = blockIdx.x * blockDim.x + threadIdx.x;
    if (i < n) p[i] = v;
}

__global__ void k_deg(const int* __restrict__ ei, int E, float* __restrict__ deg)
{
    int e = blockIdx.x * blockDim.x + threadIdx.x;
    if (e < E) atomicAdd(&deg[ei[(size_t)E + e]], 1.0f);   // dst row of edge_index
}

__global__ void k_rsqrt_inplace(float* __restrict__ p, int n)
{
    int i = blockIdx.x * blockDim.x + threadIdx.x;
    if (i < n) {
        float d = p[i];
        p[i] = d > 0.0f ? rsqrtf(d) : 0.0f;
    }
}

// agg[i,:] = hw[i,:] * dinv[i]^2   (self-loop term, also zero-initializes agg)
__global__ void k_selfloop_init(const float* __restrict__ hw,
                                const float* __restrict__ dinv,
                                float* __restrict__ agg, int Nnodes, int D)
{
    size_t idx = (size_t)blockIdx.x * blockDim.x + threadIdx.x;
    size_t total = (size_t)Nnodes * D;
    if (idx < total) {
        int i = (int)(idx / D);
        float di = dinv[i];
        agg[idx] = hw[idx] * di * di;
    }
}

// one wave per edge: agg[dst,:] += hw[src,:] * dinv[src]*dinv[dst]
__global__ __launch_bounds__(256) void k_edge_scatter(
    const float* __restrict__ hw, const int* __restrict__ ei, int E,
    const float* __restrict__ dinv, float* __restrict__ agg, int D)
{
    int gw   = (blockIdx.x * blockDim.x + threadIdx.x) >> 5;
    int lane = threadIdx.x & 31;
    if (gw >= E) return;
    int s = ei[gw];
    int d = ei[(size_t)E + gw];
    float norm = dinv[s] * dinv[d];
    const float* srow = hw + (size_t)s * D;
    float* drow = agg + (size_t)d * D;
    for (int base = lane * 4; base < D; base += 128) {
        float4 v = *reinterpret_cast<const float4*>(srow + base);
        atomicAdd(drow + base + 0, v.x * norm);
        atomicAdd(drow + base + 1, v.y * norm);
        atomicAdd(drow + base + 2, v.z * norm);
        atomicAdd(drow + base + 3, v.w * norm);
    }
}

__global__ void k_bias_relu(const float* __restrict__ agg,
                            const float* __restrict__ b,
                            float* __restrict__ h, size_t total, int Dmask)
{
    size_t idx = (size_t)blockIdx.x * blockDim.x + threadIdx.x;
    if (idx < total) {
        int j = (int)(idx & (size_t)Dmask);
        h[idx] = fmaxf(agg[idx] + b[j], 0.0f);
    }
}

// pool[batch[i], j] = max(pool, h[i, j]); h >= 0 so uint bit-ordering == float ordering
__global__ void k_pool_max(const float* __restrict__ h,
                           const int* __restrict__ batch,
                           float* __restrict__ pool, int Nnodes, int D)
{
    size_t idx = (size_t)blockIdx.x * blockDim.x + threadIdx.x;
    size_t total = (size_t)Nnodes * D;
    if (idx < total) {
        int i = (int)(idx / D);
        int j = (int)(idx - (size_t)i * D);
        unsigned bits = __float_as_uint(h[idx]);
        atomicMax((unsigned int*)&pool[(size_t)batch[i] * D + j], bits);
    }
}

// ---------------------------------------------------------------------------
extern "C" void kernel_launch(void* const* d_in, const int* in_sizes, int n_in,
                              void* d_out, int out_size, void* d_ws, size_t ws_size,
                              hipStream_t stream)
{
    const float* x     = (const float*)d_in[0];
    const int*   ei    = (const int*)d_in[1];   // [2, E] int32
    const int*   batch = (const int*)d_in[2];   // [N]    int32
    const float* W1    = (const float*)d_in[3];
    const float* b1    = (const float*)d_in[4];
    const float* W2    = (const float*)d_in[5];
    const float* b2    = (const float*)d_in[6];
    const float* fc1w  = (const float*)d_in[7];
    const float* fc1b  = (const float*)d_in[8];
    const float* fc2w  = (const float*)d_in[9];
    const float* fc2b  = (const float*)d_in[10];
    float* out = (float*)d_out;

    const int N  = in_sizes[2];        // 100000 nodes
    const int E  = in_sizes[1] / 2;    // 1.6M edges
    const int IN = 128, H = 128, H2 = 256, F1 = 512, P = 128, G = 128;

    float* ws    = (float*)d_ws;
    float* dinv  = ws;                                     // N floats (deg -> dinv)
    float* pool  = dinv + (((size_t)N + 63) & ~(size_t)63);
    float* fcmid = pool + (size_t)G * H2;                  // 128x512
    float* bufA  = fcmid + (size_t)G * F1;                 // N x 256
    float* bufB  = bufA + (size_t)N * H2;                  // N x 256

    const int TPB = 256;
    auto gemm_grid = [](int M, int Ncol) {
        int tilesM = (M + 15) / 16;
        return dim3((unsigned)(Ncol / 64), (unsigned)((tilesM + 7) / 8), 1);
    };

    // ---- degree / normalization ----
    k_fill<<<(N + TPB - 1) / TPB, TPB, 0, stream>>>(dinv, N, 1.0f);        // self-loop
    k_deg<<<(E + TPB - 1) / TPB, TPB, 0, stream>>>(ei, E, dinv);
    k_rsqrt_inplace<<<(N + TPB - 1) / TPB, TPB, 0, stream>>>(dinv, N);

    // ---- layer 1: hw1 = x @ W1 -> bufA ----
    k_gemm_f32_wmma<false><<<gemm_grid(N, H), TPB, 0, stream>>>(
        x, W1, nullptr, bufA, N, IN, H);
    // agg1 = selfloop + scatter -> bufB
    {
        size_t tot = (size_t)N * H;
        k_selfloop_init<<<(tot + TPB - 1) / TPB, TPB, 0, stream>>>(bufA, dinv, bufB, N, H);
        k_edge_scatter<<<(E + 7) / 8, TPB, 0, stream>>>(bufA, ei, E, dinv, bufB, H);
        // h1 = relu(agg1 + b1) -> bufA
        k_bias_relu<<<(tot + TPB - 1) / TPB, TPB, 0, stream>>>(bufB, b1, bufA, tot, H - 1);
    }

    // ---- layer 2: hw2 = h1 @ W2 -> bufB ----
    k_gemm_f32_wmma<false><<<gemm_grid(N, H2), TPB, 0, stream>>>(
        bufA, W2, nullptr, bufB, N, H, H2);
    // agg2 = selfloop + scatter -> bufA
    {
        size_t tot = (size_t)N * H2;
        k_selfloop_init<<<(tot + TPB - 1) / TPB, TPB, 0, stream>>>(bufB, dinv, bufA, N, H2);
        k_edge_scatter<<<(E + 7) / 8, TPB, 0, stream>>>(bufB, ei, E, dinv, bufA, H2);
        // h2 = relu(agg2 + b2) -> bufB
        k_bias_relu<<<(tot + TPB - 1) / TPB, TPB, 0, stream>>>(bufA, b2, bufB, tot, H2 - 1);
    }

    // ---- global max pool (relu output >= 0 => uint atomicMax with 0 init) ----
    {
        int pn = G * H2;
        k_fill<<<(pn + TPB - 1) / TPB, TPB, 0, stream>>>(pool, pn, 0.0f);
        size_t tot = (size_t)N * H2;
        k_pool_max<<<(tot + TPB - 1) / TPB, TPB, 0, stream>>>(bufB, batch, pool, N, H2);
    }

    // ---- FC head ----
    k_gemm_f32_wmma<true><<<gemm_grid(G, F1), TPB, 0, stream>>>(
        pool, fc1w, fc1b, fcmid, G, H2, F1);
    k_gemm_f32_wmma<true><<<gemm_grid(G, P), TPB, 0, stream>>>(
        fcmid, fc2w, fc2b, out, G, F1, P);
}